// IVDFadd8x_24318104830594
// MI455X (gfx1250) — compile-verified
//
#include <hip/hip_runtime.h>

// Problem constants from the reference
#define B_DIM 4
#define Z_DIM 6
#define Y_DIM 200
#define X_DIM 176
#define C_DIM 64
#define K_DIM (B_DIM * Z_DIM * Y_DIM * X_DIM)   // 844800 voxel slots

// ---- monotone order-preserving fp32 <-> u32 encoding --------------------
// flip(x) is strictly increasing in x, and flip(finite) >= 0x00800000,
// so 0u is a safe "empty voxel" sentinel (matches the reference's occ mask).
__device__ __forceinline__ unsigned vx_flip_f32(float x) {
    unsigned u = __float_as_uint(x);
    return (u & 0x80000000u) ? ~u : (u | 0x80000000u);
}
__device__ __forceinline__ float vx_unflip_f32(unsigned u) {
    return __uint_as_float((u & 0x80000000u) ? (u ^ 0x80000000u) : ~u);
}

// ---- 1) zero-init output (sentinel 0u everywhere), uint4 (b128) stores --
__global__ void vxfuse_init_kernel(unsigned* __restrict__ out, int n4) {
    int i = blockIdx.x * blockDim.x + threadIdx.x;
    if (i < n4) {
        ((uint4*)out)[i] = make_uint4(0u, 0u, 0u, 0u);
    }
}

// ---- 2) scatter-max of feats2 into encoded output -----------------------
// 16 threads per row, each handling 4 consecutive channels via float4 load.
__global__ void vxfuse_scatter_max_kernel(const float* __restrict__ feats,
                                          const int* __restrict__ coords,
                                          unsigned* __restrict__ out,
                                          int n_rows) {
    int t   = blockIdx.x * blockDim.x + threadIdx.x;
    int row = t >> 4;              // 16 threads / row
    int g   = t & 15;              // float4 group within the 64 channels
    if (row >= n_rows) return;

    // Prefetch a future row of the feature stream (gfx1250 global_prefetch_b8).
    if (row + 64 < n_rows) {
        __builtin_prefetch(feats + (size_t)(row + 64) * C_DIM, 0, 1);
    }

    const int4 cc = ((const int4*)coords)[row];   // [b,z,y,x], 16B aligned
    int key = ((cc.x * Z_DIM + cc.y) * Y_DIM + cc.z) * X_DIM + cc.w;

    float4 f = ((const float4*)(feats + (size_t)row * C_DIM))[g];
    unsigned* dst = out + (size_t)key * C_DIM + (g << 2);
    atomicMax(dst + 0, vx_flip_f32(f.x));
    atomicMax(dst + 1, vx_flip_f32(f.y));
    atomicMax(dst + 2, vx_flip_f32(f.z));
    atomicMax(dst + 3, vx_flip_f32(f.w));
}

// ---- 3) decode in place: sentinel -> 0.0f, else inverse transform -------
__global__ void vxfuse_decode_kernel(unsigned* __restrict__ buf, int n4) {
    int i = blockIdx.x * blockDim.x + threadIdx.x;
    if (i < n4) {
        uint4 u = ((uint4*)buf)[i];
        float4 f;
        f.x = u.x ? vx_unflip_f32(u.x) : 0.0f;
        f.y = u.y ? vx_unflip_f32(u.y) : 0.0f;
        f.z = u.z ? vx_unflip_f32(u.z) : 0.0f;
        f.w = u.w ? vx_unflip_f32(u.w) : 0.0f;
        ((float4*)buf)[i] = f;
    }
}

// ---- 4) scatter-add of feats1 on top (fused = summed1 + pooled) ---------
__global__ void vxfuse_scatter_add_kernel(const float* __restrict__ feats,
                                          const int* __restrict__ coords,
                                          float* __restrict__ out,
                                          int n_rows) {
    int t   = blockIdx.x * blockDim.x + threadIdx.x;
    int row = t >> 4;
    int g   = t & 15;
    if (row >= n_rows) return;

    if (row + 64 < n_rows) {
        __builtin_prefetch(feats + (size_t)(row + 64) * C_DIM, 0, 1);
    }

    const int4 cc = ((const int4*)coords)[row];
    int key = ((cc.x * Z_DIM + cc.y) * Y_DIM + cc.z) * X_DIM + cc.w;

    float4 f = ((const float4*)(feats + (size_t)row * C_DIM))[g];
    float* dst = out + (size_t)key * C_DIM + (g << 2);
    atomicAdd(dst + 0, f.x);   // global_atomic_add_f32
    atomicAdd(dst + 1, f.y);
    atomicAdd(dst + 2, f.z);
    atomicAdd(dst + 3, f.w);
}

extern "C" void kernel_launch(void* const* d_in, const int* in_sizes, int n_in,
                              void* d_out, int out_size, void* d_ws, size_t ws_size,
                              hipStream_t stream) {
    const float* feats1  = (const float*)d_in[0];
    const int*   coords1 = (const int*)d_in[1];
    const float* feats2  = (const float*)d_in[2];
    const int*   coords2 = (const int*)d_in[3];
    float*       out     = (float*)d_out;

    const int n1 = in_sizes[0] / C_DIM;   // 200000
    const int n2 = in_sizes[2] / C_DIM;   // 800000
    const int kc = K_DIM * C_DIM;         // == out_size (54,067,200 floats)
    const int n4 = kc / 4;

    const int TB = 256;

    // 1) sentinel-init the output
    vxfuse_init_kernel<<<(n4 + TB - 1) / TB, TB, 0, stream>>>((unsigned*)out, n4);

    // 2) encoded atomic-max pooling of the pseudo-image features
    {
        long long threads = (long long)n2 * 16;
        int blocks = (int)((threads + TB - 1) / TB);
        vxfuse_scatter_max_kernel<<<blocks, TB, 0, stream>>>(feats2, coords2,
                                                             (unsigned*)out, n2);
    }

    // 3) decode in place (empty voxels -> 0.0f)
    vxfuse_decode_kernel<<<(n4 + TB - 1) / TB, TB, 0, stream>>>((unsigned*)out, n4);

    // 4) atomic fp32 add of the sparse conv features
    {
        long long threads = (long long)n1 * 16;
        int blocks = (int)((threads + TB - 1) / TB);
        vxfuse_scatter_add_kernel<<<blocks, TB, 0, stream>>>(feats1, coords1,
                                                             out, n1);
    }
}